// TokenWiseModulatedLinear_42004780155171
// MI455X (gfx1250) — compile-verified
//
#include <hip/hip_runtime.h>
#include <hip/hip_bf16.h>

typedef float v2f __attribute__((ext_vector_type(2)));
typedef float v8f __attribute__((ext_vector_type(8)));

#define Bdim 32
#define Ndim 256
#define Cdim 512
#define Odim 512
#define Sdim 512

// ---------------------------------------------------------------------------
// Kernel 1: s_map[b,n] = (1/sqrt(S)) * sum_s s[b,s] * s_w[n,s] + s_b[n]
// 8192 dot-products of length 512 -> negligible vs. main GEMM.
// ---------------------------------------------------------------------------
__global__ __launch_bounds__(256) void smap_kernel(
    const float* __restrict__ s, const float* __restrict__ s_w,
    const float* __restrict__ s_b, float* __restrict__ smap) {
  const int idx = blockIdx.x * 256 + threadIdx.x;  // B*N = 8192
  const int b = idx >> 8;                          // / Ndim
  const int n = idx & (Ndim - 1);
  const float4* sv = reinterpret_cast<const float4*>(s + (size_t)b * Sdim);
  const float4* wv = reinterpret_cast<const float4*>(s_w + (size_t)n * Sdim);
  float acc = 0.f;
#pragma unroll 4
  for (int i = 0; i < Sdim / 4; ++i) {
    const float4 a = sv[i];
    const float4 w = wv[i];
    acc += a.x * w.x + a.y * w.y + a.z * w.z + a.w * w.w;
  }
  const float gain = 0.04419417382415922f;  // 1/sqrt(512)
  smap[idx] = acc * gain + s_b[n];
}

// ---------------------------------------------------------------------------
// Epilogue for one 16x16 D tile: out = smap * Z * rsqrt(smap^2*sumsq+eps)+bias
// D layout: VGPR v, lanes 0-15 -> M=v, lanes 16-31 -> M=v+8; N = lane&15.
// ---------------------------------------------------------------------------
__device__ __forceinline__ void store_tile(const v8f& dacc, float full,
                                           float sm, const float* __restrict__ bias_n,
                                           float* __restrict__ orow, int obase,
                                           int hh) {
  float vals[8];
#pragma unroll
  for (int v = 0; v < 8; ++v) {
    const float ss = __shfl(full, (hh << 3) + v, 32);  // row sumsq broadcast
    const float de = __builtin_amdgcn_rsqf(sm * sm * ss + 1e-8f);
    vals[v] = dacc[v] * sm * de + bias_n[obase + v];
  }
  *reinterpret_cast<float4*>(orow + obase) =
      make_float4(vals[0], vals[1], vals[2], vals[3]);
  *reinterpret_cast<float4*>(orow + obase + 4) =
      make_float4(vals[4], vals[5], vals[6], vals[7]);
}

// ---------------------------------------------------------------------------
// Kernel 2: one workgroup (8 waves) per table n.
// Each wave: 2 row-tiles of 32(O) x 32(B), K=512 via f32 WMMA 16x16x4.
// sumsq of weight rows is fused into the A-operand stream (weight read once).
// ---------------------------------------------------------------------------
__global__ __launch_bounds__(256) void modlin_kernel(
    const float* __restrict__ x, const float* __restrict__ weight,
    const float* __restrict__ bias, const float* __restrict__ smap,
    float* __restrict__ out) {
  extern __shared__ float xs[];  // Bdim * Cdim floats, XOR-swizzled (64 KB)
  const int n = blockIdx.x;
  const int tid = threadIdx.x;
  const int lane = tid & 31;
  const int wave = tid >> 5;
  const int ll = lane & 15;  // row-within-tile / b-within-tile
  const int hh = lane >> 4;  // half-wave (k sub-slot)

  // Stage x[:, n, :] into LDS. Swizzle c ^ (4*(b&15)) => conflict-free B reads.
  for (int i = tid; i < Bdim * (Cdim / 4); i += 256) {
    const int b = i >> 7;          // Cdim/4 == 128
    const int j = (i & 127) << 2;  // float offset within row
    const float4 v = *reinterpret_cast<const float4*>(
        x + ((size_t)b * Ndim + n) * Cdim + j);
    *reinterpret_cast<float4*>(xs + b * Cdim + (j ^ ((b & 15) << 2))) = v;
  }
  __syncthreads();

  const int m = ll << 2;  // LDS xor-swizzle mask (same for b and b+16)
  const int d = hh << 1;  // k sub-offset: lanes 0-15 -> K{0,1}, 16-31 -> K{2,3}
  const float* xrow0 = xs + ll * Cdim;         // b-tile 0 row
  const float* xrow1 = xs + (ll + 16) * Cdim;  // b-tile 1 row
  const float* bias_n = bias + (size_t)n * Odim;
  const float sm0 = smap[(size_t)ll * Ndim + n];
  const float sm1 = smap[(size_t)(ll + 16) * Ndim + n];
  float* orow0 = out + ((size_t)ll * Ndim + n) * Odim;
  float* orow1 = out + ((size_t)(ll + 16) * Ndim + n) * Odim;

  for (int t = 0; t < 2; ++t) {
    const int o0 = (wave * 2 + t) * 32;
    // A operand base: lane l -> row (o0 + l&15), k offset 2*(l>>4)
    const float* wp0 = weight + ((size_t)n * Odim + o0 + ll) * Cdim + d;
    const float* wp1 = wp0 + 16 * Cdim;
    v8f acc00 = {}, acc01 = {}, acc10 = {}, acc11 = {};
    float ssq0 = 0.f, ssq1 = 0.f;
#pragma unroll 4
    for (int k = 0; k < Cdim; k += 4) {
      const int kb = (k ^ m) + d;
      const v2f a0 = *(const v2f*)(wp0 + k);    // global_load_b64
      const v2f a1 = *(const v2f*)(wp1 + k);
      const v2f b0 = *(const v2f*)(xrow0 + kb); // ds_load_b64
      const v2f b1 = *(const v2f*)(xrow1 + kb);
      acc00 = __builtin_amdgcn_wmma_f32_16x16x4_f32(false, a0, false, b0,
                                                    (short)0, acc00, false, false);
      acc01 = __builtin_amdgcn_wmma_f32_16x16x4_f32(false, a0, false, b1,
                                                    (short)0, acc01, false, false);
      acc10 = __builtin_amdgcn_wmma_f32_16x16x4_f32(false, a1, false, b0,
                                                    (short)0, acc10, false, false);
      acc11 = __builtin_amdgcn_wmma_f32_16x16x4_f32(false, a1, false, b1,
                                                    (short)0, acc11, false, false);
      ssq0 += a0.x * a0.x + a0.y * a0.y;  // fused sum_c weight^2 (rows o0+ll)
      ssq1 += a1.x * a1.x + a1.y * a1.y;  // rows o0+16+ll
    }
    // Combine half-wave k-slices: every lane then holds full sumsq of row l&15.
    const float full0 = ssq0 + __shfl_xor(ssq0, 16, 32);
    const float full1 = ssq1 + __shfl_xor(ssq1, 16, 32);

    const int ob0 = o0 + (hh << 3);       // a-tile 0
    const int ob1 = o0 + 16 + (hh << 3);  // a-tile 1
    store_tile(acc00, full0, sm0, bias_n, orow0, ob0, hh);
    store_tile(acc01, full0, sm1, bias_n, orow1, ob0, hh);
    store_tile(acc10, full1, sm0, bias_n, orow0, ob1, hh);
    store_tile(acc11, full1, sm1, bias_n, orow1, ob1, hh);
  }
}

// ---------------------------------------------------------------------------
extern "C" void kernel_launch(void* const* d_in, const int* in_sizes, int n_in,
                              void* d_out, int out_size, void* d_ws,
                              size_t ws_size, hipStream_t stream) {
  (void)in_sizes; (void)n_in; (void)out_size; (void)ws_size;
  const float* x    = (const float*)d_in[0];  // (B, N, C)
  const float* s    = (const float*)d_in[1];  // (B, S)
  const float* w    = (const float*)d_in[2];  // (N, O, C)
  const float* bias = (const float*)d_in[3];  // (1, N, O)
  const float* s_w  = (const float*)d_in[4];  // (N, S)
  const float* s_b  = (const float*)d_in[5];  // (N,)
  float* out  = (float*)d_out;                // (B, N, O)
  float* smap = (float*)d_ws;                 // B*N floats scratch

  smap_kernel<<<(Bdim * Ndim) / 256, 256, 0, stream>>>(s, s_w, s_b, smap);
  modlin_kernel<<<Ndim, 256, Bdim * Cdim * sizeof(float), stream>>>(x, w, bias,
                                                                    smap, out);
}